// Schweizer_65970697666735
// MI455X (gfx1250) — compile-verified
//
#include <hip/hip_runtime.h>
#include <hip/hip_bf16.h>
#include <math.h>

// CDNA5 / gfx1250 — wave32, WMMA f32 16x16x4.
//
// out[:, 0:12]    = x
// out[:, 12:793]  = 1 - (1 - prod_{i in S} w_i)^(1/lambda),  w_i = 1 - (1-x_i)^lambda
// via log-space GEMM:  S = LW (B x 12) @ IND (12 x 781),  IND = 0/1 combination masks,
// then P = exp(S), out = 1 - (1-P)^(1/lambda).

typedef __attribute__((ext_vector_type(2))) float v2f;
typedef __attribute__((ext_vector_type(8))) float v8f;

#define N_FEAT 12
#define N_COMB 781          // C(12,2)+C(12,3)+C(12,4) = 66+220+495
#define N_OUT  793          // 12 + 781
#define NTILES 49           // ceil(781/16) -> 784 padded columns
#define ROWS_PER_BLOCK 16
#define THREADS 256
#define WAVES 8

// ---------- compile-time combination-mask table (itertools.combinations order) ----
struct MaskTable { unsigned short v[NTILES * 16]; };

constexpr MaskTable make_masks() {
    MaskTable t{};
    int p = 0;
    for (int size = 2; size <= 4; ++size) {
        int c[4] = {0, 0, 0, 0};
        for (int i = 0; i < size; ++i) c[i] = i;
        for (;;) {
            unsigned short m = 0;
            for (int i = 0; i < size; ++i) m = (unsigned short)(m | (1u << c[i]));
            t.v[p++] = m;
            int i = size - 1;
            while (i >= 0 && c[i] == N_FEAT - size + i) --i;
            if (i < 0) break;
            ++c[i];
            for (int j = i + 1; j < size; ++j) c[j] = c[j - 1] + 1;
        }
    }
    for (; p < NTILES * 16; ++p) t.v[p] = 0;   // pad tiles -> zero columns
    return t;
}

__constant__ MaskTable g_masks = make_masks();

// ---------------------------------- kernel ---------------------------------------
__global__ __launch_bounds__(THREADS)
void schweizer_wmma_kernel(const float* __restrict__ x,
                           const float* __restrict__ lam_p,
                           float* __restrict__ out,
                           int rows)
{
    __shared__ float s_lw[ROWS_PER_BLOCK][N_FEAT];   // clamped log(w), 16x12

    const float lam = *lam_p;
    const float inv = 1.0f / lam;

    const int tid   = threadIdx.x;
    const int rbase = blockIdx.x * ROWS_PER_BLOCK;
    const bool fullTile = (rbase + ROWS_PER_BLOCK) <= rows;   // block-uniform

    // Stage 1: 192 threads load the 16x12 tile, write identity columns,
    // and stage lw = max(log(1-(1-x)^lam), -64) into LDS.
    if (tid < ROWS_PER_BLOCK * N_FEAT) {
        const int m   = tid / N_FEAT;
        const int k   = tid - m * N_FEAT;
        const int row = rbase + m;
        float lw = -64.0f;
        if (fullTile || row < rows) {
            const float xv = x[row * N_FEAT + k];
            out[row * N_OUT + k] = xv;                       // identity part
            const float t = powf(1.0f - xv, lam);            // (1-x)^lambda
            const float w = 1.0f - t;                        // in [0, 1)
            lw = fmaxf(logf(w), -64.0f);                     // -inf-safe clamp
        }
        s_lw[m][k] = lw;
    }
    __syncthreads();

    // Stage 2: per-wave WMMA over N-tiles.
    const int lane   = tid & 31;           // wave32
    const int wave   = tid >> 5;
    const int am     = lane & 15;          // A-matrix row M
    const int koff   = (lane >> 4) << 1;   // 0 (lanes 0-15) or 2 (lanes 16-31)
    const int nloc   = lane & 15;          // B/D column within tile
    const int rowoff = (lane >> 4) << 3;   // D rows: +0 or +8

    // A fragments (16x4 f32): VGPR0 = K=koff, VGPR1 = K=koff+1, three K-steps.
    const v2f a0 = { s_lw[am][0 + koff], s_lw[am][1 + koff] };
    const v2f a1 = { s_lw[am][4 + koff], s_lw[am][5 + koff] };
    const v2f a2 = { s_lw[am][8 + koff], s_lw[am][9 + koff] };

    // Per-lane output base for the combo block: row (rbase+rowoff), col N_FEAT.
    float* const obase = out + (size_t)(rbase + rowoff) * N_OUT + N_FEAT;

    for (int t = wave; t < NTILES; t += WAVES) {
        const int ncol = t * 16 + nloc;
        const unsigned int mask = g_masks.v[ncol];

        // B fragments (4x16 f32): lanes hold N, VGPR0 = K=koff, VGPR1 = K=koff+1.
        const v2f b0 = { (mask >> (0 + koff)) & 1u ? 1.0f : 0.0f,
                         (mask >> (1 + koff)) & 1u ? 1.0f : 0.0f };
        const v2f b1 = { (mask >> (4 + koff)) & 1u ? 1.0f : 0.0f,
                         (mask >> (5 + koff)) & 1u ? 1.0f : 0.0f };
        const v2f b2 = { (mask >> (8 + koff)) & 1u ? 1.0f : 0.0f,
                         (mask >> (9 + koff)) & 1u ? 1.0f : 0.0f };

        v8f c = {};
        c = __builtin_amdgcn_wmma_f32_16x16x4_f32(false, a0, false, b0,
                                                  (short)0, c, false, false);
        c = __builtin_amdgcn_wmma_f32_16x16x4_f32(false, a1, false, b1,
                                                  (short)0, c, false, false);
        c = __builtin_amdgcn_wmma_f32_16x16x4_f32(false, a2, false, b2,
                                                  (short)0, c, false, false);

        // Epilogue: out = 1 - (1 - exp(s))^(1/lambda); store D layout
        // (VGPR r, lane l) -> row = rbase + rowoff + r, col = ncol.
        float v[8];
        #pragma unroll
        for (int r = 0; r < 8; ++r) {
            const float s = c[r];                 // sum of logs (<= ~-0.013)
            const float P = expf(s);              // product of w_i
            v[r] = 1.0f - powf(1.0f - P, inv);
        }

        const bool fullCols = (t * 16 + 16) <= N_COMB;        // wave-uniform
        if (fullTile && fullCols) {
            // Fast path: no divergence, straight-line stores.
            #pragma unroll
            for (int r = 0; r < 8; ++r)
                obase[(size_t)r * N_OUT + ncol] = v[r];
        } else if (ncol < N_COMB) {
            #pragma unroll
            for (int r = 0; r < 8; ++r) {
                if (fullTile || (rbase + rowoff + r) < rows)
                    obase[(size_t)r * N_OUT + ncol] = v[r];
            }
        }
    }
}

// --------------------------------- launcher --------------------------------------
extern "C" void kernel_launch(void* const* d_in, const int* in_sizes, int n_in,
                              void* d_out, int out_size, void* d_ws, size_t ws_size,
                              hipStream_t stream) {
    const float* x   = (const float*)d_in[0];
    const float* lam = (const float*)d_in[1];
    float* out       = (float*)d_out;

    const int rows = in_sizes[0] / N_FEAT;                 // 32768
    const int grid = (rows + ROWS_PER_BLOCK - 1) / ROWS_PER_BLOCK;

    schweizer_wmma_kernel<<<grid, THREADS, 0, stream>>>(x, lam, out, rows);
}